// GCN_66760971649325
// MI455X (gfx1250) — compile-verified
//
#include <hip/hip_runtime.h>
#include <hip/hip_bf16.h>

typedef __attribute__((ext_vector_type(2))) float v2f;
typedef __attribute__((ext_vector_type(8))) float v8f;

constexpr int N_NODES = 100000;
constexpr int N_EDGES = 1600000;
constexpr int FEAT    = 128;
constexpr int HID     = 64;   // = OUT
constexpr int NIDX    = 10000;

// ---------------------------------------------------------------------------
// GEMM: out[N x 64] = act(X[N x K]) @ W[K x 64] + bias
// ACT: relu(x)*mask (fused GCN relu + inverted dropout), else identity.
// W staged in LDS PAIR-PACKED: sW[k/2*64 + n] = {W[k][n], W[k+1][n]} so each
// WMMA B fragment is a single aligned ds_load_b64 into an even VGPR pair
// (no v_mov marshalling in front of the WMMAs).
// One wave computes a 16x64 tile via 4 accumulators of V_WMMA_F32_16X16X4_F32.
// 10 waves / 320-thread block; grid = N/16/10 = 625 exactly -> EXEC all-1s.
// ---------------------------------------------------------------------------
template<int K, bool ACT>
__global__ __launch_bounds__(320) void gcn_gemm_wmma(
    const float* __restrict__ X,      // N x K
    const float* __restrict__ W,      // K x 64 (row-major)
    const float* __restrict__ bias,   // 64
    const float* __restrict__ mask,   // N x K (read only if ACT)
    float* __restrict__ out)          // N x 64
{
    __shared__ v2f sW[(K / 2) * 64];
    // Pair-packed staging: entry p -> (k2 = p>>6, n = p&63) = {W[2k2][n], W[2k2+1][n]}
    #pragma unroll 4
    for (int p = threadIdx.x; p < (K / 2) * 64; p += 320) {
        const int k2 = p >> 6;
        const int n  = p & 63;
        v2f t;
        t.x = W[(size_t)(2 * k2)     * 64 + n];
        t.y = W[(size_t)(2 * k2 + 1) * 64 + n];
        sW[p] = t;
    }
    __syncthreads();

    const int wave   = threadIdx.x >> 5;
    const int lane   = threadIdx.x & 31;
    const int tile   = blockIdx.x * 10 + wave;  // 16-row tile, 0..6249
    const int m0     = tile * 16;
    const int hiHalf = (lane >= 16) ? 1 : 0;
    const int l15    = lane & 15;
    const int arow   = m0 + l15;                // row this lane feeds into A
    const int koff   = hiHalf * 2;              // K sub-offset per lane half

    v8f acc[4];
    #pragma unroll
    for (int c = 0; c < 4; ++c) acc[c] = (v8f){0.f,0.f,0.f,0.f,0.f,0.f,0.f,0.f};

    const float* xrow  = X + (size_t)arow * K;
    const float* mrow  = ACT ? (mask + (size_t)arow * K) : nullptr;
    const v2f*   wbase = sW + hiHalf * 64 + l15;   // per-lane, k-invariant part of B addr

    #pragma unroll 2
    for (int k = 0; k < K; k += 4) {
        // ---- A fragment: 16x4 f32; lane holds (arow, k+koff), (arow, k+koff+1)
        v2f a = *(const v2f*)(xrow + k + koff);           // global_load_b64
        if (ACT) {
            const float* mp = mrow + k + koff;
            a.x = fmaxf(a.x, 0.0f) * mp[0];
            a.y = fmaxf(a.y, 0.0f) * mp[1];
        }

        // ---- B fragments: single ds_load_b64 each (pair row = k/2 + hiHalf)
        const v2f* wp = wbase + (size_t)(k >> 1) * 64;
        #pragma unroll
        for (int c = 0; c < 4; ++c) {
            v2f b = wp[c * 16];                           // {W[k+koff][n], W[k+koff+1][n]}
            acc[c] = __builtin_amdgcn_wmma_f32_16x16x4_f32(
                /*neg_a=*/false, a, /*neg_b=*/false, b,
                /*c_mod=*/(short)0, acc[c],
                /*reuse_a=*/false, /*reuse_b=*/false);
        }
    }

    // ---- Epilogue: C/D layout -> VGPR v is row (m0 + v + 8*hiHalf), col c*16+l15
    const int mrow16 = m0 + hiHalf * 8;
    #pragma unroll
    for (int c = 0; c < 4; ++c) {
        const int n    = c * 16 + l15;
        const float bv = bias[n];
        #pragma unroll
        for (int v = 0; v < 8; ++v) {
            out[(size_t)(mrow16 + v) * 64 + n] = acc[c][v] + bv;
        }
    }
}

// ---------------------------------------------------------------------------
// SpMM scatter: out[row[e]][:] += vals[e] * H[col[e]][:]
// 16 threads per edge, float4 lanes -> 256B coalesced global_load_b128 gather;
// f32 atomic scatter resolves in the 192MB L2 (H is 25.6MB, fully resident).
// Grid is exact (E*16 threads), no bounds check.
// ---------------------------------------------------------------------------
__global__ __launch_bounds__(256) void gcn_spmm_atomic(
    const float* __restrict__ vals, const int* __restrict__ row,
    const int* __restrict__ col, const float* __restrict__ H,
    float* __restrict__ out)
{
    const long long gid = (long long)blockIdx.x * 256 + threadIdx.x;
    const int e = (int)(gid >> 4);
    const int j = (int)(gid & 15) << 2;
    const float v = vals[e];
    const int c = col[e];
    const int r = row[e];
    const float4 h = *(const float4*)(H + (size_t)c * 64 + j);
    float* o = out + (size_t)r * 64 + j;
    atomicAdd(o + 0, v * h.x);
    atomicAdd(o + 1, v * h.y);
    atomicAdd(o + 2, v * h.z);
    atomicAdd(o + 3, v * h.w);
}

__global__ __launch_bounds__(256) void gcn_zero4(float4* __restrict__ p)
{
    const long long i = (long long)blockIdx.x * 256 + threadIdx.x;
    p[i] = make_float4(0.f, 0.f, 0.f, 0.f);
}

__global__ __launch_bounds__(256) void gcn_gather(
    const float* __restrict__ H, const int* __restrict__ idx,
    float* __restrict__ out)
{
    const int t = blockIdx.x * 256 + threadIdx.x;   // NIDX*16 threads exactly
    const int i = t >> 4;
    const int j = (t & 15) << 2;
    *(float4*)(out + (size_t)i * 64 + j) =
        *(const float4*)(H + (size_t)idx[i] * 64 + j);
}

// ---------------------------------------------------------------------------
extern "C" void kernel_launch(void* const* d_in, const int* in_sizes, int n_in,
                              void* d_out, int out_size, void* d_ws, size_t ws_size,
                              hipStream_t stream)
{
    const float* x    = (const float*)d_in[0];
    const float* vals = (const float*)d_in[1];
    const float* W1   = (const float*)d_in[2];
    const float* b1   = (const float*)d_in[3];
    const float* W2   = (const float*)d_in[4];
    const float* b2   = (const float*)d_in[5];
    const float* mask = (const float*)d_in[6];
    const int*   row  = (const int*)d_in[7];
    const int*   col  = (const int*)d_in[8];
    const int*   idx  = (const int*)d_in[9];
    float* outp = (float*)d_out;

    float* bufA = (float*)d_ws;                       // N x 64
    float* bufB = bufA + (size_t)N_NODES * HID;       // N x 64

    const int gemmBlocks   = N_NODES / 16 / 10;                         // 625
    const int zeroBlocks   = (int)((long long)N_NODES * HID / 4 / 256); // 6250
    const int spmmBlocks   = (int)((long long)N_EDGES * 16 / 256);      // 100000
    const int gatherBlocks = NIDX * 16 / 256;                           // 625

    // Layer 1: bufA = x @ W1 + b1
    gcn_gemm_wmma<FEAT, false><<<gemmBlocks, 320, 0, stream>>>(x, W1, b1, nullptr, bufA);
    // bufB = A_hat @ bufA
    gcn_zero4<<<zeroBlocks, 256, 0, stream>>>((float4*)bufB);
    gcn_spmm_atomic<<<spmmBlocks, 256, 0, stream>>>(vals, row, col, bufA, bufB);
    // Layer 2: bufA = (relu(bufB)*mask) @ W2 + b2   (activation fused into A load)
    gcn_gemm_wmma<HID, true><<<gemmBlocks, 320, 0, stream>>>(bufB, W2, b2, mask, bufA);
    // bufB = A_hat @ bufA
    gcn_zero4<<<zeroBlocks, 256, 0, stream>>>((float4*)bufB);
    gcn_spmm_atomic<<<spmmBlocks, 256, 0, stream>>>(vals, row, col, bufA, bufB);
    // out = bufB[idx]
    gcn_gather<<<gatherBlocks, 256, 0, stream>>>(bufB, idx, outp);
}